// ChannelwiseTensorProduct_63728724738604
// MI455X (gfx1250) — compile-verified
//
#include <hip/hip_runtime.h>
#include <stdint.h>

#define MULC 32
#define INV_SQRT3f 0.57735026918962576451f

// ---------------------------------------------------------------------------
// CDNA5 async global->LDS copy path (gfx1250). Each wave copies its own 512B
// weight row into its private LDS slice; ASYNCcnt is per-wave state, so no
// block barrier is required. Builtin signature (from clang diagnostic):
//   void __builtin_amdgcn_global_load_async_to_lds_b128(
//       int4 AS(1)* gaddr, int4 AS(3)* ldsaddr, int offset, int cpol)
// ---------------------------------------------------------------------------
#if __has_builtin(__builtin_amdgcn_global_load_async_to_lds_b128) && \
    __has_builtin(__builtin_amdgcn_s_wait_asynccnt)
#define USE_ASYNC_LDS 1
typedef int v4i_t __attribute__((ext_vector_type(4)));
typedef __attribute__((address_space(1))) v4i_t* g1_v4i_p;
typedef __attribute__((address_space(3))) v4i_t* l3_v4i_p;
#else
#define USE_ASYNC_LDS 0
#endif

// ---------------------------------------------------------------------------
// Edge kernel: one wave32 per edge, lane == channel (MUL = 32).
// Defined FIRST so the disasm snippet shows this kernel's body.
// ---------------------------------------------------------------------------
__global__ void __launch_bounds__(256) ctp_edges(
    const float* __restrict__ weights,   // (E, 4, 32)
    const float* __restrict__ x1_0e,     // (N, 32, 1)
    const float* __restrict__ x1_1o,     // (N, 32, 3)
    const float* __restrict__ x2_0e,     // (E, 1, 1)
    const float* __restrict__ x2_1o,     // (E, 1, 3)
    const int*   __restrict__ senders,   // (E,)
    const int*   __restrict__ receivers, // (E,)
    float* __restrict__ out0,            // (N, 64, 1)
    float* __restrict__ out1,            // (N, 64, 3)
    int num_edges)
{
    const int lane = threadIdx.x & 31;
    const int wid  = threadIdx.x >> 5;                 // wave in block: 0..7
    const int e    = blockIdx.x * 8 + wid;             // one edge per wave32

#if USE_ASYNC_LDS
    __shared__ float wsm[8 * 128];                     // 512B per wave
    if (e < num_edges) {
        const float* gsrc = weights + (size_t)e * 128 + lane * 4;   // 16B/lane
        float*       ldst = &wsm[wid * 128 + lane * 4];
        // AS(1) global src; AS(3) LDS dst via integer round-trip (low 32 bits
        // of a generic LDS address are the LDS offset).
        __builtin_amdgcn_global_load_async_to_lds_b128(
            (g1_v4i_p)(uintptr_t)gsrc,
            (l3_v4i_p)(uint32_t)(uintptr_t)ldst,
            /*offset=*/0, /*cpol=*/0);
    }
#endif
    if (e >= num_edges) return;

    // Prefetch next block's weight tile (streaming, read-once).
    __builtin_prefetch(weights + ((size_t)e + 8) * 128 + lane * 4, 0, 0);

    const int s = senders[e];
    const int r = receivers[e];

    // Per-edge scalars: uniform across the wave (hardware merges the loads).
    const float sh0 = __builtin_nontemporal_load(&x2_0e[e]);
    const float b0  = __builtin_nontemporal_load(&x2_1o[(size_t)e * 3 + 0]);
    const float b1  = __builtin_nontemporal_load(&x2_1o[(size_t)e * 3 + 1]);
    const float b2  = __builtin_nontemporal_load(&x2_1o[(size_t)e * 3 + 2]);

    // Gathered sender features: lane == channel. x1 is only 25.6MB -> stays
    // resident in the 192MB L2 across its ~16x reuse; use regular (RT) loads.
    const float s0  = x1_0e[(size_t)s * MULC + lane];
    const float* s1p = x1_1o + (size_t)s * (MULC * 3) + lane * 3;
    const float a0 = s1p[0];
    const float a1 = s1p[1];
    const float a2 = s1p[2];

    float w0, w1, w2, w3;
#if USE_ASYNC_LDS
    __builtin_amdgcn_s_wait_asynccnt(0);
    {
        const float* wp = &wsm[wid * 128];
        w0 = wp[0 * MULC + lane];
        w1 = wp[1 * MULC + lane];
        w2 = wp[2 * MULC + lane];
        w3 = wp[3 * MULC + lane];
    }
#else
    {
        const float* wp = weights + (size_t)e * 128;   // streaming: NT hint
        w0 = __builtin_nontemporal_load(wp + 0 * MULC + lane);
        w1 = __builtin_nontemporal_load(wp + 1 * MULC + lane);
        w2 = __builtin_nontemporal_load(wp + 2 * MULC + lane);
        w3 = __builtin_nontemporal_load(wp + 3 * MULC + lane);
    }
#endif

    // Channelwise CG tensor product (EPSILON == 1).
    const float dot  = a0 * b0 + a1 * b1 + a2 * b2;
    const float m0a  = w0 * (s0 * sh0);                // 0e x 0e -> 0e
    const float m0b  = w3 * (dot * INV_SQRT3f);        // 1o x 1o -> 0e
    const float m1a0 = w1 * (s0 * b0);                 // 0e x 1o -> 1o
    const float m1a1 = w1 * (s0 * b1);
    const float m1a2 = w1 * (s0 * b2);
    const float m1b0 = w2 * (a0 * sh0);                // 1o x 0e -> 1o
    const float m1b1 = w2 * (a1 * sh0);
    const float m1b2 = w2 * (a2 * sh0);

    // Scatter-add to receiver. Output (51MB) is L2-resident; fp32 atomics
    // resolve at L2 (native global_atomic_add_f32).
    float* o0 = out0 + (size_t)r * (2 * MULC);
    atomicAdd(o0 + lane,        m0a);
    atomicAdd(o0 + MULC + lane, m0b);

    float* o1  = out1 + (size_t)r * (2 * MULC * 3);
    float* o1a = o1 + lane * 3;
    atomicAdd(o1a + 0, m1a0);
    atomicAdd(o1a + 1, m1a1);
    atomicAdd(o1a + 2, m1a2);
    float* o1b = o1 + (MULC + lane) * 3;
    atomicAdd(o1b + 0, m1b0);
    atomicAdd(o1b + 1, m1b1);
    atomicAdd(o1b + 2, m1b2);
}

// ---------------------------------------------------------------------------
// Zero-fill d_out (harness poisons it; atomics above accumulate into zeros).
// ---------------------------------------------------------------------------
__global__ void __launch_bounds__(256) zero_kernel(float* __restrict__ p, int n) {
    const int n4 = n >> 2;
    float4* p4 = (float4*)p;
    for (int i = blockIdx.x * blockDim.x + threadIdx.x; i < n4;
         i += gridDim.x * blockDim.x) {
        p4[i] = float4{0.f, 0.f, 0.f, 0.f};
    }
    if (blockIdx.x == 0 && threadIdx.x < (n & 3)) {
        p[(n4 << 2) + threadIdx.x] = 0.f;
    }
}

// ---------------------------------------------------------------------------
// Launch
// ---------------------------------------------------------------------------
extern "C" void kernel_launch(void* const* d_in, const int* in_sizes, int n_in,
                              void* d_out, int out_size, void* d_ws, size_t ws_size,
                              hipStream_t stream) {
    const float* weights   = (const float*)d_in[0];
    const float* x1_0e     = (const float*)d_in[1];
    const float* x1_1o     = (const float*)d_in[2];
    const float* x2_0e     = (const float*)d_in[3];
    const float* x2_1o     = (const float*)d_in[4];
    const int*   senders   = (const int*)d_in[5];
    const int*   receivers = (const int*)d_in[6];

    const int E = in_sizes[0] / 128;   // (E, 4*32)
    const int N = in_sizes[1] / 32;    // (N, 32, 1)

    float* out0 = (float*)d_out;
    float* out1 = out0 + (size_t)N * 64;

    // 1) zero the output accumulators
    {
        const int n4 = out_size >> 2;
        int blocks = (n4 + 255) / 256;
        if (blocks > 65535) blocks = 65535;
        if (blocks < 1) blocks = 1;
        zero_kernel<<<blocks, 256, 0, stream>>>((float*)d_out, out_size);
    }

    // 2) edge messages + atomic scatter (one wave32 per edge, lane = channel)
    {
        const int blocks = (E + 7) / 8;
        ctp_edges<<<blocks, 256, 0, stream>>>(weights, x1_0e, x1_1o, x2_0e, x2_1o,
                                              senders, receivers, out0, out1, E);
    }
}